// GraphPairClassifier_79620103733766
// MI455X (gfx1250) — compile-verified
//
#include <hip/hip_runtime.h>
#include <math.h>

typedef __attribute__((ext_vector_type(2))) float v2f;
typedef __attribute__((ext_vector_type(8))) float v8f;

#define HID 128
#define NGRAPH 2048

// ---------------------------------------------------------------- utilities

__global__ void k_fill(float* __restrict__ p, long n, float val) {
  long i = (long)blockIdx.x * blockDim.x + threadIdx.x;
  long stride = (long)gridDim.x * blockDim.x;
  for (; i < n; i += stride) p[i] = val;
}

// degree count with self loop already in (caller fills with 1.0f first)
__global__ void k_deg(const int* __restrict__ dst, float* __restrict__ deg, int E) {
  int e = blockIdx.x * blockDim.x + threadIdx.x;
  if (e < E) unsafeAtomicAdd(&deg[dst[e]], 1.0f);
}

__global__ void k_rsqrt(float* __restrict__ d, int n) {
  int i = blockIdx.x * blockDim.x + threadIdx.x;
  if (i < n) d[i] = rsqrtf(d[i]);
}

// ---------------------------------------------------------------- GEMM K=3 (layer 1)
// x is raw input features (no activation before the first conv).

__global__ void k_gemm_in3(const float* __restrict__ x, const float* __restrict__ W,
                           float* __restrict__ out, int N) {
  long idx = (long)blockIdx.x * blockDim.x + threadIdx.x;
  if (idx >= (long)N * HID) return;
  int n = (int)(idx >> 7), o = (int)(idx & 127);
  float s = x[n * 3 + 0] * W[0 * HID + o]
          + x[n * 3 + 1] * W[1 * HID + o]
          + x[n * 3 + 2] * W[2 * HID + o];
  out[idx] = s;
}

// ---------------------------------------------------------------- GEMM 128x128 via f32 WMMA
// Block = 256 threads = 8 waves. Block stages a 16-row strip of A in LDS,
// applying the deferred ReLU of the previous layer during the staging load.
// Wave w owns output column tile [w*16, w*16+16); K loop = 32 x V_WMMA_F32_16X16X4_F32.
// Requires N % 16 == 0 (N = 100000 -> 6250 blocks, exact).

__global__ void __launch_bounds__(256)
k_gemm128_relu(const float* __restrict__ A, const float* __restrict__ W,
               float* __restrict__ Out, int N) {
  __shared__ float As[16 * HID];
  const int tid = threadIdx.x;
  const long m0 = (long)blockIdx.x * 16;

  // cooperative load of 16x128 strip with fused ReLU (prev layer's activation)
  #pragma unroll
  for (int i = 0; i < 8; ++i) {
    int idx = tid + i * 256;
    As[idx] = fmaxf(A[m0 * HID + idx], 0.0f);
  }
  __syncthreads();

  const int wave = tid >> 5;
  const int lane = tid & 31;
  const int half = lane >> 4;   // 0: lanes 0-15, 1: lanes 16-31
  const int r    = lane & 15;
  const int n0   = wave * 16;

  v8f acc = {};
  #pragma unroll 4
  for (int k0 = 0; k0 < HID; k0 += 4) {
    const int ka = k0 + 2 * half;
    // A-matrix 16x4 f32: VGPR0 = K(+0 | +2), VGPR1 = K(+1 | +3), M = lane&15
    v2f a;
    a.x = As[r * HID + ka];
    a.y = As[r * HID + ka + 1];
    // B-matrix 4x16 f32: rows striped across lanes within a VGPR (mirror of A)
    v2f b;
    b.x = W[(ka + 0) * HID + n0 + r];
    b.y = W[(ka + 1) * HID + n0 + r];
    acc = __builtin_amdgcn_wmma_f32_16x16x4_f32(
        /*neg_a=*/false, a, /*neg_b=*/false, b,
        /*c_mod=*/(short)0, acc, /*reuse_a=*/false, /*reuse_b=*/false);
  }

  // C/D 16x16 f32: VGPR j -> M = j + 8*half, N = lane&15
  #pragma unroll
  for (int j = 0; j < 8; ++j) {
    Out[(m0 + j + 8 * half) * HID + n0 + r] = acc[j];
  }
}

// ---------------------------------------------------------------- self-term init
// agg = t * dis[n]^2 + bias  (replaces zero-fill + post-combine pass)

__global__ void k_selfinit(const float* __restrict__ t, const float* __restrict__ dis,
                           const float* __restrict__ bias, float* __restrict__ agg,
                           int N) {
  long idx = (long)blockIdx.x * blockDim.x + threadIdx.x;
  if (idx >= (long)N * HID) return;
  int n = (int)(idx >> 7), o = (int)(idx & 127);
  float sn = dis[n] * dis[n];
  agg[idx] = t[idx] * sn + bias[o];
}

// ---------------------------------------------------------------- edge aggregation
// One wave per edge; lane owns a float4 chunk of the 128 features.
// t (N x 128 = 51 MB) stays L2-resident, so random gathers hit L2.

__global__ void k_aggregate(const int* __restrict__ src, const int* __restrict__ dst,
                            const float* __restrict__ dis, const float* __restrict__ t,
                            float* __restrict__ agg, int E) {
  long gid = (long)blockIdx.x * blockDim.x + threadIdx.x;
  int e = (int)(gid >> 5);
  if (e >= E) return;
  int lane = (int)(gid & 31);
  int s = src[e], d = dst[e];
  float norm = dis[s] * dis[d];
  const float4 v = *(const float4*)(t + (long)s * HID + lane * 4);
  float* ap = agg + (long)d * HID + lane * 4;
  unsafeAtomicAdd(ap + 0, v.x * norm);
  unsafeAtomicAdd(ap + 1, v.y * norm);
  unsafeAtomicAdd(ap + 2, v.z * norm);
  unsafeAtomicAdd(ap + 3, v.w * norm);
}

// ---------------------------------------------------------------- mean pool (atomic sums)
// Applies the deferred ReLU of the final layer on load.

__global__ void k_pool(const float* __restrict__ h, const int* __restrict__ batch,
                       float* __restrict__ pooled, float* __restrict__ cnts,
                       int N, int base) {
  long gid = (long)blockIdx.x * blockDim.x + threadIdx.x;
  int n = (int)(gid >> 5);
  if (n >= N) return;
  int lane = (int)(gid & 31);
  int g = batch[n];
  const float4 v = *(const float4*)(h + (long)n * HID + lane * 4);
  float* pp = pooled + (long)g * 256 + base + lane * 4;
  unsafeAtomicAdd(pp + 0, fmaxf(v.x, 0.0f));
  unsafeAtomicAdd(pp + 1, fmaxf(v.y, 0.0f));
  unsafeAtomicAdd(pp + 2, fmaxf(v.z, 0.0f));
  unsafeAtomicAdd(pp + 3, fmaxf(v.w, 0.0f));
  if (lane == 0) unsafeAtomicAdd(&cnts[g], 1.0f);
}

// ---------------------------------------------------------------- per-graph MLP head
// One 128-thread block per graph; LDS ping-pong through 256->128->64->32->16->1.

__global__ void __launch_bounds__(128)
k_mlp(const float* __restrict__ pooled,
      const float* __restrict__ c1, const float* __restrict__ c2,
      const float* __restrict__ W0, const float* __restrict__ b0,
      const float* __restrict__ W1, const float* __restrict__ b1,
      const float* __restrict__ W2, const float* __restrict__ b2,
      const float* __restrict__ W3, const float* __restrict__ b3,
      const float* __restrict__ W4, const float* __restrict__ b4,
      float* __restrict__ out) {
  __shared__ float A[256];
  __shared__ float B[128];
  const int g = blockIdx.x;
  const int t = threadIdx.x;
  const float inv1 = 1.0f / fmaxf(c1[g], 1.0f);
  const float inv2 = 1.0f / fmaxf(c2[g], 1.0f);
  A[t]       = pooled[(long)g * 256 + t] * inv1;
  A[t + 128] = pooled[(long)g * 256 + 128 + t] * inv2;
  __syncthreads();
  { // 256 -> 128
    float s = b0[t];
    for (int k = 0; k < 256; ++k) s += A[k] * W0[k * 128 + t];
    B[t] = fmaxf(s, 0.0f);
  }
  __syncthreads();
  if (t < 64) { // 128 -> 64
    float s = b1[t];
    for (int k = 0; k < 128; ++k) s += B[k] * W1[k * 64 + t];
    A[t] = fmaxf(s, 0.0f);
  }
  __syncthreads();
  if (t < 32) { // 64 -> 32
    float s = b2[t];
    for (int k = 0; k < 64; ++k) s += A[k] * W2[k * 32 + t];
    B[t] = fmaxf(s, 0.0f);
  }
  __syncthreads();
  if (t < 16) { // 32 -> 16
    float s = b3[t];
    for (int k = 0; k < 32; ++k) s += B[k] * W3[k * 16 + t];
    A[t] = fmaxf(s, 0.0f);
  }
  __syncthreads();
  if (t == 0) { // 16 -> 1, sigmoid
    float s = b4[0];
    for (int k = 0; k < 16; ++k) s += A[k] * W4[k];
    out[g] = 1.0f / (1.0f + expf(-s));
  }
}

// ---------------------------------------------------------------- driver

extern "C" void kernel_launch(void* const* d_in, const int* in_sizes, int n_in,
                              void* d_out, int out_size, void* d_ws, size_t ws_size,
                              hipStream_t stream) {
  const float* x1  = (const float*)d_in[0];
  const float* x2  = (const float*)d_in[1];
  const int*   ei1 = (const int*)d_in[2];
  const int*   ei2 = (const int*)d_in[3];
  const int*   bt1 = (const int*)d_in[4];
  const int*   bt2 = (const int*)d_in[5];
  const float* cw[5]; const float* cb[5];
  for (int i = 0; i < 5; ++i) { cw[i] = (const float*)d_in[6 + 2 * i];  cb[i] = (const float*)d_in[7 + 2 * i]; }
  const float* mw[5]; const float* mb[5];
  for (int i = 0; i < 5; ++i) { mw[i] = (const float*)d_in[16 + 2 * i]; mb[i] = (const float*)d_in[17 + 2 * i]; }

  const int N = in_sizes[4];      // batch_1 length
  const int E = in_sizes[2] / 2;  // edge_index rows are (2, E)

  float* ws     = (float*)d_ws;
  float* bufA   = ws;                               // N*128 (ping)
  float* bufB   = bufA + (size_t)N * HID;           // N*128 (pong)
  float* t      = bufB + (size_t)N * HID;           // N*128 (pre-activation h@W)
  float* dis    = t   + (size_t)N * HID;            // N
  float* pooled = dis + (size_t)N;                  // G*256
  float* cnt1   = pooled + (size_t)NGRAPH * 256;    // G
  float* cnt2   = cnt1 + NGRAPH;                    // G
  float* outp   = (float*)d_out;

  const long NH   = (long)N * HID;
  const int bEdge = (int)((((long)E << 5) + 255) / 256);
  const int bNH   = (int)((NH + 255) / 256);
  const int bPool = (int)((((long)N << 5) + 255) / 256);
  const int bGemm = N / 16;

  // zero pooled sums + both counts (contiguous)
  k_fill<<<2048, 256, 0, stream>>>(pooled, (long)NGRAPH * 256 + 2 * NGRAPH, 0.0f);

  const float* xs[2]  = { x1, x2 };
  const int*   eis[2] = { ei1, ei2 };
  const int*   bts[2] = { bt1, bt2 };
  float*       cns[2] = { cnt1, cnt2 };

  for (int enc = 0; enc < 2; ++enc) {
    const float* x   = xs[enc];
    const int*   src = eis[enc];
    const int*   dst = eis[enc] + E;
    // deg = 1 (self loop) + count(dst); dis = rsqrt(deg)
    k_fill<<<512, 256, 0, stream>>>(dis, (long)N, 1.0f);
    k_deg<<<(E + 255) / 256, 256, 0, stream>>>(dst, dis, E);
    k_rsqrt<<<(N + 255) / 256, 256, 0, stream>>>(dis, N);

    // Layer state: prev = pre-ReLU result of previous layer (ReLU fused into
    // consumers); ping-pongs between bufA and bufB.
    float* prev = bufA;
    float* next = bufB;
    for (int layer = 0; layer < 5; ++layer) {
      if (layer == 0)
        k_gemm_in3<<<bNH, 256, 0, stream>>>(x, cw[0], t, N);
      else
        k_gemm128_relu<<<bGemm, 256, 0, stream>>>(prev, cw[layer], t, N);
      // next = t*self_norm + bias, then scatter-add edge contributions
      k_selfinit<<<bNH, 256, 0, stream>>>(t, dis, cb[layer], next, N);
      k_aggregate<<<bEdge, 256, 0, stream>>>(src, dst, dis, t, next, E);
      float* tmp = prev; prev = next; next = tmp;
    }
    // prev now holds the pre-ReLU layer-5 output; ReLU fused into pooling
    k_pool<<<bPool, 256, 0, stream>>>(prev, bts[enc], pooled, cns[enc], N, enc * HID);
  }

  k_mlp<<<NGRAPH, 128, 0, stream>>>(pooled, cnt1, cnt2,
                                    mw[0], mb[0], mw[1], mb[1], mw[2], mb[2],
                                    mw[3], mb[3], mw[4], mb[4], outp);
}